// GaussianModel_35287451304162
// MI455X (gfx1250) — compile-verified
//
#include <hip/hip_runtime.h>
#include <cstdint>
#include <cstddef>

#define BLOCK 256
#define FR_FLOATS 45   // features_rest: 15 coeffs * 3 channels per gaussian

typedef uint32_t u32x4 __attribute__((ext_vector_type(4)));
typedef uint32_t u32x8 __attribute__((ext_vector_type(8)));

// SH constants (match reference)
#define C0f  0.28209479177387814f
#define C1f  0.4886025119029199f
#define C2_0 1.0925484305920792f
#define C2_1 (-1.0925484305920792f)
#define C2_2 0.31539156525252005f
#define C2_3 (-1.0925484305920792f)
#define C2_4 0.5462742152960396f
#define C3_0 (-0.5900435899266435f)
#define C3_1 2.890611442640554f
#define C3_2 (-0.4570457994644658f)
#define C3_3 0.3731763325901154f
#define C3_4 (-0.4570457994644658f)
#define C3_5 1.445305721320277f
#define C3_6 (-0.5900435899266435f)

__global__ __launch_bounds__(BLOCK) void gaussian_cov_sh_kernel(
    const float* __restrict__ scaling_logits,  // N*3
    const float* __restrict__ rotation,        // N*4  (w,x,y,z)
    const float* __restrict__ features_dc,     // N*3
    const float* __restrict__ features_rest,   // N*45 (15 coeffs x 3 ch)
    const float* __restrict__ view_dirs,       // N*3
    float* __restrict__ out_cov,               // N*9
    float* __restrict__ out_col,               // N*3
    int N)
{
    // 45 KB: first the features_rest stage, then (after barrier) reused to
    // stage output slabs for coalesced async stores.
    __shared__ float lds_buf[BLOCK * FR_FLOATS];

    const int tid   = threadIdx.x;
    const int base  = blockIdx.x * BLOCK;
    const int count = min(BLOCK, N - base);       // >= 1 for every launched block

    // ---- Stage features_rest slab into LDS via the Tensor Data Mover ----
    // Each of the 8 waves DMAs its 1/8 of the contiguous count*45-float slab
    // with ONE tensor_load_to_lds (TENSORcnt-tracked), instead of ~90
    // per-lane copy instructions. All-wave split avoids the "EXEC ignored"
    // duplicate-issue hazard of a single-issuer branch.
    {
        const int nFloats = count * FR_FLOATS;            // <= 11520
        const int wave    = tid >> 5;                     // wave32: 8 waves
        const int cpw     = (nFloats + 7) >> 3;           // floats per wave
        const int beg     = wave * cpw;
        int cnt           = nFloats - beg;
        cnt = (cnt < 0) ? 0 : ((cnt > cpw) ? cpw : cnt);  // tile_dim0==0 -> NOP

        const float*  src   = features_rest + (size_t)base * FR_FLOATS + beg;
        const uint64_t gaddr = (uint64_t)(uintptr_t)src;
        // Low 32 bits of a generic LDS pointer == offset in the WG allocation.
        const uint32_t laddr = (uint32_t)(uintptr_t)(&lds_buf[beg]);

        // D# group 0: {count=1 | lds_addr | global_addr[56:0] | type=2}
        u32x4 g0;
        g0[0] = 1u;                                        // count=1, user mode
        g0[1] = laddr;                                     // lds_addr (bytes)
        g0[2] = (uint32_t)gaddr;                           // global_addr[31:0]
        g0[3] = ((uint32_t)(gaddr >> 32) & 0x01FFFFFFu)    // global_addr[56:32]
              | (2u << 30);                                // type=2 ("image")

        // D# group 1: 1-D tensor, data_size=4B, dims/tile/stride = cnt
        u32x8 g1;
        g1[0] = 2u << 16;                                  // wg_mask=0, data_size=4B
        g1[1] = ((uint32_t)cnt & 0xFFFFu) << 16;           // tensor_dim0[15:0]
        g1[2] = ((uint32_t)cnt >> 16) | (1u << 16);        // tensor_dim0[31:16], tensor_dim1=1
        g1[3] = ((uint32_t)cnt & 0xFFFFu) << 16;           // tensor_dim1 hi=0, tile_dim0=cnt
        g1[4] = 1u;                                        // tile_dim1=1, tile_dim2=0
        g1[5] = (uint32_t)cnt;                             // tensor_dim0_stride[31:0]
        g1[6] = 0u;                                        // stride0 hi, stride1 lo
        g1[7] = 0u;                                        // stride1 hi

        asm volatile("tensor_load_to_lds %0, %1" :: "s"(g0), "s"(g1) : "memory");
#if defined(__has_builtin) && __has_builtin(__builtin_amdgcn_s_wait_tensorcnt)
        __builtin_amdgcn_s_wait_tensorcnt(0);
#else
        asm volatile("s_wait_tensorcnt 0x0" ::: "memory");
#endif
    }
    __syncthreads();

    float c00 = 0.f, c01 = 0.f, c02 = 0.f, c11 = 0.f, c12 = 0.f, c22 = 0.f;
    float r = 0.f, g = 0.f, b = 0.f;

    if (tid < count) {
        const int n = base + tid;

        // ---------------- Covariance ----------------
        const float sl0 = scaling_logits[(size_t)n * 3 + 0];
        const float sl1 = scaling_logits[(size_t)n * 3 + 1];
        const float sl2 = scaling_logits[(size_t)n * 3 + 2];
        const float sx = __expf(sl0);
        const float sy = __expf(sl1);
        const float sz = __expf(sl2);

        const float4 q = *(const float4*)(rotation + (size_t)n * 4); // 16B aligned
        const float qn  = q.x*q.x + q.y*q.y + q.z*q.z + q.w*q.w;
        const float inv = 1.0f / fmaxf(sqrtf(qn), 1e-12f);
        const float w = q.x * inv, x = q.y * inv, y = q.z * inv, z = q.w * inv;

        const float ww = w*w, xx_ = x*x, yy_ = y*y, zz_ = z*z;
        const float wx = w*x, wy = w*y, wz = w*z;
        const float xy_ = x*y, xz_ = x*z, yz_ = y*z;

        // R rows, then M = R * diag(s)
        const float m00 = (ww + xx_ - yy_ - zz_) * sx;
        const float m01 = (2.0f * (xy_ - wz))    * sy;
        const float m02 = (2.0f * (xz_ + wy))    * sz;
        const float m10 = (2.0f * (xy_ + wz))    * sx;
        const float m11 = (ww - xx_ + yy_ - zz_) * sy;
        const float m12 = (2.0f * (yz_ - wx))    * sz;
        const float m20 = (2.0f * (xz_ - wy))    * sx;
        const float m21 = (2.0f * (yz_ + wx))    * sy;
        const float m22 = (ww - xx_ - yy_ + zz_) * sz;

        c00 = m00*m00 + m01*m01 + m02*m02;
        c01 = m00*m10 + m01*m11 + m02*m12;
        c02 = m00*m20 + m01*m21 + m02*m22;
        c11 = m10*m10 + m11*m11 + m12*m12;
        c12 = m10*m20 + m11*m21 + m12*m22;
        c22 = m20*m20 + m21*m21 + m22*m22;

        // ---------------- SH degree 3 ----------------
        const float dx = view_dirs[(size_t)n * 3 + 0];
        const float dy = view_dirs[(size_t)n * 3 + 1];
        const float dz = view_dirs[(size_t)n * 3 + 2];
        const float xx = dx*dx, yy = dy*dy, zz = dz*dz;
        const float xy = dx*dy, yz = dy*dz, xz = dx*dz;

        float bs[15];
        bs[0]  = -C1f * dy;
        bs[1]  =  C1f * dz;
        bs[2]  = -C1f * dx;
        bs[3]  =  C2_0 * xy;
        bs[4]  =  C2_1 * yz;
        bs[5]  =  C2_2 * (2.0f * zz - xx - yy);
        bs[6]  =  C2_3 * xz;
        bs[7]  =  C2_4 * (xx - yy);
        bs[8]  =  C3_0 * dy * (3.0f * xx - yy);
        bs[9]  =  C3_1 * xy * dz;
        bs[10] =  C3_2 * dy * (4.0f * zz - xx - yy);
        bs[11] =  C3_3 * dz * (2.0f * zz - 3.0f * xx - 3.0f * yy);
        bs[12] =  C3_4 * dx * (4.0f * zz - xx - yy);
        bs[13] =  C3_5 * dz * (xx - yy);
        bs[14] =  C3_6 * dx * (xx - yy - 3.0f * zz);

        r = C0f * features_dc[(size_t)n * 3 + 0];
        g = C0f * features_dc[(size_t)n * 3 + 1];
        b = C0f * features_dc[(size_t)n * 3 + 2];

        const float* fr = &lds_buf[tid * FR_FLOATS];  // stride 45: conflict-free
        #pragma unroll
        for (int k = 0; k < 15; ++k) {
            r = fmaf(bs[k], fr[k * 3 + 0], r);
            g = fmaf(bs[k], fr[k * 3 + 1], g);
            b = fmaf(bs[k], fr[k * 3 + 2], b);
        }

        r = fminf(fmaxf(r + 0.5f, 0.0f), 1.0f);
        g = fminf(fmaxf(g + 0.5f, 0.0f), 1.0f);
        b = fminf(fmaxf(b + 0.5f, 0.0f), 1.0f);
    }

    // ---- Reuse LDS for output staging (results now live in registers) ----
    __syncthreads();                              // all feature reads done

    float* lds_cov = lds_buf;                     // count*9 floats  (<= 9 KB)
    float* lds_col = lds_buf + BLOCK * 9;         // count*3 floats; 16B-aligned base

    if (tid < count) {
        float* oc = lds_cov + tid * 9;            // stride 9: conflict-free
        oc[0] = c00; oc[1] = c01; oc[2] = c02;
        oc[3] = c01; oc[4] = c11; oc[5] = c12;
        oc[6] = c02; oc[7] = c12; oc[8] = c22;
        float* ocl = lds_col + tid * 3;           // stride 3: conflict-free
        ocl[0] = r; ocl[1] = g; ocl[2] = b;
    }
    __syncthreads();                              // slabs complete in LDS

    // ---- Drain to global with coalesced async b128 streaming stores ----
    // Output slabs are contiguous per block; NT hint keeps the 96 MB
    // write-once stream from evicting useful L2 lines. s_endpgm's implicit
    // wait-idle guarantees the DMA drains before LDS is released.
    {
        const int nF = count * 9;
        const int n4 = nF >> 2;
        float* dst = out_cov + (size_t)base * 9;
        for (int c = tid; c < n4; c += BLOCK) {
            float* gp = dst + (size_t)c * 4;
            uint32_t l = (uint32_t)(uintptr_t)(&lds_cov[c * 4]);
            asm volatile("global_store_async_from_lds_b128 %0, %1, off th:TH_STORE_NT"
                         :: "v"(gp), "v"(l) : "memory");
        }
        const int rem = nF & 3;
        if (tid < rem) dst[(size_t)n4 * 4 + tid] = lds_cov[n4 * 4 + tid];
    }
    {
        const int nF = count * 3;
        const int n4 = nF >> 2;
        float* dst = out_col + (size_t)base * 3;
        for (int c = tid; c < n4; c += BLOCK) {
            float* gp = dst + (size_t)c * 4;
            uint32_t l = (uint32_t)(uintptr_t)(&lds_col[c * 4]);
            asm volatile("global_store_async_from_lds_b128 %0, %1, off th:TH_STORE_NT"
                         :: "v"(gp), "v"(l) : "memory");
        }
        const int rem = nF & 3;
        if (tid < rem) dst[(size_t)n4 * 4 + tid] = lds_col[n4 * 4 + tid];
    }
}

extern "C" void kernel_launch(void* const* d_in, const int* in_sizes, int n_in,
                              void* d_out, int out_size, void* d_ws, size_t ws_size,
                              hipStream_t stream) {
    (void)n_in; (void)d_ws; (void)ws_size; (void)out_size;
    const int N = in_sizes[0] / 3;                       // scaling_logits is (N,3)
    const float* scaling_logits = (const float*)d_in[0];
    const float* rotation       = (const float*)d_in[1];
    const float* features_dc    = (const float*)d_in[2];
    const float* features_rest  = (const float*)d_in[3];
    const float* view_dirs      = (const float*)d_in[4];

    float* out_cov = (float*)d_out;                      // N*9, then colors N*3
    float* out_col = out_cov + (size_t)N * 9;

    const int grid = (N + BLOCK - 1) / BLOCK;
    gaussian_cov_sh_kernel<<<grid, BLOCK, 0, stream>>>(
        scaling_logits, rotation, features_dc, features_rest, view_dirs,
        out_cov, out_col, N);
}